// sat_layer3d_31078383354176
// MI455X (gfx1250) — compile-verified
//
#include <hip/hip_runtime.h>

typedef __attribute__((ext_vector_type(2))) float v2f;
typedef __attribute__((ext_vector_type(4))) float v4f;
typedef __attribute__((ext_vector_type(8))) float v8f;

// geometry constants
constexpr int HWn       = 3136;      // 56*56 plane
constexpr int Wn        = 56;
constexpr int PPn       = 64;        // padded plane row pitch (16B-aligned interior)
constexpr int PSZ       = 3712;      // 58*64 padded plane
constexpr int NCH       = 256;
constexpr int NDe       = 16;
constexpr int CH_STRIDE = 50176;     // 16*3136
constexpr int B_STRIDE  = 12845056;  // 256*50176
// interior pixel (y,x) stored at row y+1, col x+4 (col is multiple of 4 -> b128 OK)

// LDS float-offset of conv tap k relative to base (y*64 + x + 3)
__device__ __forceinline__ int tap_off(int k) {
  int c = k / 27, r = k % 27;
  return (c * 3 + r / 9) * PSZ + ((r % 9) / 3) * PPn + (r % 3);
}
// fast sigmoid: v_exp_f32 + v_add + v_rcp_f32 (no IEEE divide expansion)
__device__ __forceinline__ float sigmoidf(float v) {
  return __builtin_amdgcn_rcpf(1.0f + __expf(-v));
}
// channel shuffle: pre-shuffle channel cp -> final channel
__device__ __forceinline__ int shuf(int cp) { return ((cp & 127) << 1) | (cp >> 7); }

extern "C" __global__ void __launch_bounds__(256)
sat_layer3d_fused(const float* __restrict__ x, const float* __restrict__ cweight,
                  const float* __restrict__ cbias, const float* __restrict__ conv_w,
                  float* __restrict__ out)
{
  extern __shared__ float smem[];
  float* s_xs  = smem;             // 6*3712 = 22272 : [mean|max][dz 0..2][58][64] zero-padded
  float* s_x0  = s_xs + 22272;     // 2*3136
  float* s_x1  = s_x0 + 6272;      // 2*3136
  float* s_w   = s_x1 + 6272;      // 56 (54 taps + 2 zero pads)
  float* s_sum = s_w + 56;         // 2

  const int tid  = threadIdx.x;
  const int bgd  = blockIdx.x;
  const int dpos = bgd & 15;       // d slice
  const int bg   = bgd >> 4;       // group-batch 0..255
  const int b    = bg >> 6;
  const int g    = bg & 63;

  // ---- phase 0: zero padded xs planes (b128), load weights ----
  for (int i = tid; i < 5568; i += 256) { v4f z = {}; *(v4f*)(s_xs + i * 4) = z; }
  if (tid < 56) s_w[tid] = (tid < 54) ? conv_w[tid] : 0.0f;
  if (tid < 2)  s_sum[tid] = 0.0f;
  __syncthreads();

  // ---- phase 1: stage x0, x1(d), mean/max xs planes (d-1,d,d+1); all b128 ----
  const float* p0     = x + (size_t)b * B_STRIDE + (size_t)(4 * g) * CH_STRIDE
                          + (size_t)dpos * HWn;
  const float* p2base = x + (size_t)b * B_STRIDE + (size_t)(4 * g + 2) * CH_STRIDE;
  float sum0 = 0.0f, sum1 = 0.0f;
  for (int q4 = tid; q4 < 784; q4 += 256) {      // 784 float4-chunks per plane
    int p = q4 * 4;                              // row-aligned: 56 % 4 == 0
    v4f a0 = *(const v4f*)(p0 + p);
    v4f a1 = *(const v4f*)(p0 + CH_STRIDE + p);
    *(v4f*)(s_x0 + p)       = a0;
    *(v4f*)(s_x0 + HWn + p) = a1;
    sum0 += a0.x + a0.y + a0.z + a0.w;
    sum1 += a1.x + a1.y + a1.z + a1.w;
    int y = p / Wn, xx = p - y * Wn;
    int xb = (y + 1) * PPn + (xx + 4);           // 16B-aligned interior col
#pragma unroll
    for (int dz = 0; dz < 3; ++dz) {
      int dd = dpos + dz - 1;
      if (dd >= 0 && dd < NDe) {                 // out-of-range d stays zero (SAME pad)
        const float* q = p2base + (size_t)dd * HWn;
        v4f u = *(const v4f*)(q + p);            // halo re-reads hit 192MB L2
        v4f v = *(const v4f*)(q + CH_STRIDE + p);
        v4f mn = (u + v) * 0.5f;
        v4f mx; mx.x = fmaxf(u.x, v.x); mx.y = fmaxf(u.y, v.y);
                mx.z = fmaxf(u.z, v.z); mx.w = fmaxf(u.w, v.w);
        *(v4f*)(s_xs + dz * PSZ + xb)       = mn;   // xs ch0 = mean
        *(v4f*)(s_xs + (3 + dz) * PSZ + xb) = mx;   // xs ch1 = max
        if (dz == 1) { *(v4f*)(s_x1 + p) = u; *(v4f*)(s_x1 + HWn + p) = v; }
      }
    }
  }
  atomicAdd(&s_sum[0], sum0);   // ds_add_f32
  atomicAdd(&s_sum[1], sum1);
  __syncthreads();

  // ---- phase 2: xn = x0 * sigmoid(cweight*mean + cbias); NT b128 stores ----
  {
    float m0 = s_sum[0] * (1.0f / HWn);
    float m1 = s_sum[1] * (1.0f / HWn);
    float g0 = sigmoidf(cweight[dpos]      * m0 + cbias[dpos]);
    float g1 = sigmoidf(cweight[16 + dpos] * m1 + cbias[16 + dpos]);
    float* o0 = out + ((size_t)b * NCH + shuf(4 * g))     * CH_STRIDE + (size_t)dpos * HWn;
    float* o1 = out + ((size_t)b * NCH + shuf(4 * g + 1)) * CH_STRIDE + (size_t)dpos * HWn;
    for (int q4 = tid; q4 < 784; q4 += 256) {
      int p = q4 * 4;
      v4f a0 = *(const v4f*)(s_x0 + p);
      v4f a1 = *(const v4f*)(s_x0 + HWn + p);
      __builtin_nontemporal_store(a0 * g0, (v4f*)(o0 + p));
      __builtin_nontemporal_store(a1 * g1, (v4f*)(o1 + p));
    }
  }

  // ---- phase 3: 3x3x3 conv via V_WMMA_F32_16X16X4_F32, fused x_out store ----
  // A(16x4) = weights replicated over M rows; B(4x16) = 16 pixels' patches as columns.
  // D[m,n] = conv(pixel n) replicated over m -> lane l holds pixel (l&15) in acc[0];
  // lane halves 0/1 write output channels 0/1 directly.
  {
    const int lane = tid & 31;
    const int wid  = tid >> 5;
    const int half = lane >> 4;       // K-group select AND output channel
    const int n    = lane & 15;       // pixel-in-tile (B column)

    // hoist j-invariant A-operand weights and tap offsets into registers.
    // padded taps (54,55) get zero WEIGHT; their B slot reads a valid dummy tap.
    float wa0[14], wa1[14];
    int   offx[14], offy[14];
#pragma unroll
    for (int s = 0; s < 14; ++s) {
      const int k0 = 4 * s;
      int kx = half ? k0 + 2 : k0;
      int ky = half ? k0 + 3 : k0 + 1;
      offx[s] = tap_off(kx >= 54 ? 52 : kx);
      offy[s] = tap_off(ky >= 54 ? 53 : ky);
      wa0[s]  = (kx >= 54) ? 0.0f : s_w[kx];
      wa1[s]  = (ky >= 54) ? 0.0f : s_w[ky];
    }

    float* oA  = out + ((size_t)b * NCH + shuf(4 * g + 2)) * CH_STRIDE + (size_t)dpos * HWn;
    float* oB  = out + ((size_t)b * NCH + shuf(4 * g + 3)) * CH_STRIDE + (size_t)dpos * HWn;
    float* och = half ? oB : oA;

    for (int j = wid; j < 196; j += 8) {     // 196 tiles of 16 pixels; wave-uniform
      int p  = j * 16 + n;
      int y  = p / Wn, xx = p - y * Wn;
      const float* bp = s_xs + (y * PPn + xx + 3);  // tap (dy,dx) adds dy*64+dx
      v8f acc = {};
#pragma unroll
      for (int s = 0; s < 14; ++s) {
        v2f A;  A.x = wa0[s];       A.y = wa1[s];
        v2f Bv; Bv.x = bp[offx[s]]; Bv.y = bp[offy[s]];
        acc = __builtin_amdgcn_wmma_f32_16x16x4_f32(false, A, false, Bv,
                                                    (short)0, acc, false, false);
      }
      float cv = acc[0];                      // conv result for pixel n, any row
      float ov = sigmoidf(cv) * s_x1[half * HWn + p];
      __builtin_nontemporal_store(ov, och + p);
    }
  }
}

extern "C" void kernel_launch(void* const* d_in, const int* in_sizes, int n_in,
                              void* d_out, int out_size, void* d_ws, size_t ws_size,
                              hipStream_t stream) {
  (void)in_sizes; (void)n_in; (void)out_size; (void)d_ws; (void)ws_size;
  const float* x       = (const float*)d_in[0];
  const float* cweight = (const float*)d_in[1];
  const float* cbias   = (const float*)d_in[2];
  const float* conv_w  = (const float*)d_in[3];
  float* out = (float*)d_out;

  const size_t shmem = (size_t)(22272 + 6272 + 6272 + 56 + 2) * sizeof(float); // 139,496 B
  hipFuncSetAttribute((const void*)sat_layer3d_fused,
                      hipFuncAttributeMaxDynamicSharedMemorySize, (int)shmem);
  // 256 group-batches * 16 d-slices = 4096 blocks, 256 threads (8 wave32) each
  sat_layer3d_fused<<<4096, 256, shmem, stream>>>(x, cweight, cbias, conv_w, out);
}